// TripletLossWithHardMining_74766790689431
// MI455X (gfx1250) — compile-verified
//
#include <hip/hip_runtime.h>
#include <hip/hip_bf16.h>
#include <math.h>

// Problem constants: B=256, N=256, D=512
#define DB 256
#define DN 256
#define DD 512
#define ND (DN * DD)          // 131072 floats per batch row slice
#define NPAIR (DB * DN)       // 65536 (i,j) pairs
#define EPS_F 1e-6f
#define MARGIN_F 0.5f

typedef __attribute__((ext_vector_type(2))) float v2f;
typedef __attribute__((ext_vector_type(8))) float v8f;

// ---------------------------------------------------------------------------
// Phase 1: per-pair squared-L2 reduction over D=512.
// One wave handles 16 consecutive pairs. Lane L: pair r = L&15, half h = L>>4.
// Per-lane FMA partials land exactly in the V_WMMA_F32_16X16X4_F32 A-matrix
// layout (lanes 0-15 = K0/K1, lanes 16-31 = K2/K3), so a single fp32 WMMA
// against an all-ones B collapses each pair's 4 partials; D-matrix layout puts
// pair sums in c[0..7] of lanes 0 and 16.
// ---------------------------------------------------------------------------
__global__ __launch_bounds__(256) void phase1_dist(
    const float* __restrict__ A, const float* __restrict__ P,
    const float* __restrict__ Ng, float* __restrict__ dp, float* __restrict__ dn)
{
  const int tid  = threadIdx.x;
  const int lane = tid & 31;
  const int wv   = tid >> 5;
  const int r    = lane & 15;
  const int h    = lane >> 4;
  const long p0  = (long)blockIdx.x * 128 + (long)wv * 16;   // first pair of this wave
  const size_t row = (size_t)(p0 + r) * DD;

  const float4* a4 = (const float4*)(A  + row);
  const float4* p4 = (const float4*)(P  + row);
  const float4* n4 = (const float4*)(Ng + row);

  float apx = 0.f, apy = 0.f;   // dist_pos partials (k0/k2, k1/k3 slots)
  float anx = 0.f, any_ = 0.f;  // dist_neg partials

#pragma unroll 4
  for (int d = 0; d < DD / 4; d += 2) {
    const int idx = d + h;                    // halves interleave float4 slots
    const float4 a = a4[idx];
    const float4 p = p4[idx];
    const float4 n = n4[idx];
    float t;
    t = a.x - p.x + EPS_F; apx = fmaf(t, t, apx);
    t = a.y - p.y + EPS_F; apy = fmaf(t, t, apy);
    t = a.z - p.z + EPS_F; apx = fmaf(t, t, apx);
    t = a.w - p.w + EPS_F; apy = fmaf(t, t, apy);
    t = a.x - n.x + EPS_F; anx = fmaf(t, t, anx);
    t = a.y - n.y + EPS_F; any_ = fmaf(t, t, any_);
    t = a.z - n.z + EPS_F; anx = fmaf(t, t, anx);
    t = a.w - n.w + EPS_F; any_ = fmaf(t, t, any_);
  }

  v2f Ap;  Ap.x = apx;  Ap.y = apy;
  v2f An;  An.x = anx;  An.y = any_;
  v2f ones; ones.x = 1.0f; ones.y = 1.0f;   // all-ones B: layout-independent
  v8f z = {};

  // D[m][n] = sum_k A[m][k]  (identical across columns n)
  v8f cp = __builtin_amdgcn_wmma_f32_16x16x4_f32(
      false, Ap, false, ones, (short)0, z, false, false);
  v8f cn = __builtin_amdgcn_wmma_f32_16x16x4_f32(
      false, An, false, ones, (short)0, z, false, false);

  // D layout: c[v] @ lane 0 -> pair v ; c[v] @ lane 16 -> pair v+8
  if (lane == 0) {
#pragma unroll
    for (int v = 0; v < 8; ++v) {
      dp[p0 + v] = sqrtf(cp[v]);
      dn[p0 + v] = sqrtf(cn[v]);
    }
  } else if (lane == 16) {
#pragma unroll
    for (int v = 0; v < 8; ++v) {
      dp[p0 + 8 + v] = sqrtf(cp[v]);
      dn[p0 + 8 + v] = sqrtf(cn[v]);
    }
  }
}

// ---------------------------------------------------------------------------
// Phase 2: per-row mining. Exact first-index tie-breaking to match
// jnp.argmax/argmin semantics. Invalid rows encode index -1.
// ---------------------------------------------------------------------------
__global__ __launch_bounds__(256) void phase2_mine(
    const float* __restrict__ dp, const float* __restrict__ dn,
    int* __restrict__ negidx, int* __restrict__ posidx,
    float* __restrict__ rowloss, float* __restrict__ out_vn,
    float* __restrict__ out_vp)
{
  __shared__ float sv[256];
  __shared__ int   si[256];
  __shared__ float sl[256];

  const int i = blockIdx.x;
  const int j = threadIdx.x;
  const float a = dp[i * DN + j];
  const float b = dn[i * DN + j];

  sl[j] = fmaxf(0.0f, MARGIN_F + a - b);

  // hard-negative: among j with dp>dn, argmax of dn (first max on ties)
  sv[j] = (a > b) ? b : -__builtin_inff();
  si[j] = j;
  __syncthreads();
  for (int s = 128; s > 0; s >>= 1) {
    if (j < s) {
      const float v2 = sv[j + s];
      const int   i2 = si[j + s];
      if (v2 > sv[j] || (v2 == sv[j] && i2 < si[j])) { sv[j] = v2; si[j] = i2; }
      sl[j] += sl[j + s];
    }
    __syncthreads();
  }
  if (j == 0) {
    const bool valid = sv[0] > -__builtin_inff();
    negidx[i]  = valid ? si[0] : -1;
    out_vn[i]  = valid ? 1.0f : 0.0f;
    rowloss[i] = sl[0];
  }
  __syncthreads();

  // hard-positive: among j with dn>dp, argmin of dp (first min on ties)
  sv[j] = (b > a) ? a : __builtin_inff();
  si[j] = j;
  __syncthreads();
  for (int s = 128; s > 0; s >>= 1) {
    if (j < s) {
      const float v2 = sv[j + s];
      const int   i2 = si[j + s];
      if (v2 < sv[j] || (v2 == sv[j] && i2 < si[j])) { sv[j] = v2; si[j] = i2; }
    }
    __syncthreads();
  }
  if (j == 0) {
    const bool valid = sv[0] < __builtin_inff();
    posidx[i] = valid ? si[0] : -1;
    out_vp[i] = valid ? 1.0f : 0.0f;
  }
}

// ---------------------------------------------------------------------------
// Deterministic final loss reduction (256 row partials -> mean).
// ---------------------------------------------------------------------------
__global__ __launch_bounds__(256) void loss_reduce(
    const float* __restrict__ rowloss, float* __restrict__ out_loss)
{
  __shared__ float s[256];
  const int t = threadIdx.x;
  s[t] = rowloss[t];
  __syncthreads();
  for (int k = 128; k > 0; k >>= 1) {
    if (t < k) s[t] += s[t + k];
    __syncthreads();
  }
  if (t == 0) out_loss[0] = s[0] * (1.0f / (float)NPAIR);
}

// ---------------------------------------------------------------------------
// Phase 3: gather-copy negative[neg_idx[i]] / positive[pos_idx[i]] (or zeros).
// d_out base is offset by 1 float, so use fully-coalesced dword accesses
// (lanes write consecutive dwords; 4 elements per thread at stride 256).
// grid = (128 chunks of 1024 floats, 256 rows).
// ---------------------------------------------------------------------------
__global__ __launch_bounds__(256) void phase3_gather(
    const float* __restrict__ Pos, const float* __restrict__ Neg,
    const int* __restrict__ negidx, const int* __restrict__ posidx,
    float* __restrict__ out_hn, float* __restrict__ out_hp)
{
  const int i    = blockIdx.y;
  const int t    = threadIdx.x;
  const int base = blockIdx.x * 1024;
  const int ni = negidx[i];
  const int pi = posidx[i];
  const size_t rowoff = (size_t)i * ND;
  const float* nsrc = Neg + (size_t)(ni < 0 ? 0 : ni) * ND;
  const float* psrc = Pos + (size_t)(pi < 0 ? 0 : pi) * ND;
#pragma unroll
  for (int c = 0; c < 4; ++c) {
    const int e = base + c * 256 + t;
    out_hn[rowoff + e] = (ni >= 0) ? nsrc[e] : 0.0f;
    out_hp[rowoff + e] = (pi >= 0) ? psrc[e] : 0.0f;
  }
}

extern "C" void kernel_launch(void* const* d_in, const int* in_sizes, int n_in,
                              void* d_out, int out_size, void* d_ws, size_t ws_size,
                              hipStream_t stream) {
  (void)in_sizes; (void)n_in; (void)out_size; (void)ws_size;
  const float* anchor   = (const float*)d_in[0];
  const float* positive = (const float*)d_in[1];
  const float* negative = (const float*)d_in[2];

  float* out = (float*)d_out;
  const size_t BND = (size_t)DB * DN * DD;   // 33,554,432 elements per tensor out
  float* out_loss = out;
  float* out_hn   = out + 1;
  float* out_hp   = out + 1 + BND;
  float* out_vn   = out + 1 + 2 * BND;
  float* out_vp   = out_vn + DB;

  // dp/dn scratch lives in the hard_neg output region (phase3 fully
  // overwrites it afterwards); only ~3KB of d_ws is needed.
  float* dp = out_hn;                 // NPAIR floats
  float* dn = out_hn + NPAIR;        // NPAIR floats
  float* rowloss = (float*)d_ws;                           // 256 floats
  int*   negidx  = (int*)((char*)d_ws + 256 * sizeof(float));  // 256 ints
  int*   posidx  = negidx + DB;                            // 256 ints

  phase1_dist<<<NPAIR / 128, 256, 0, stream>>>(anchor, positive, negative, dp, dn);
  phase2_mine<<<DB, 256, 0, stream>>>(dp, dn, negidx, posidx, rowloss, out_vn, out_vp);
  loss_reduce<<<1, 256, 0, stream>>>(rowloss, out_loss);
  phase3_gather<<<dim3(ND / 1024, DB), 256, 0, stream>>>(positive, negative,
                                                         negidx, posidx, out_hn, out_hp);
}